// GraphConvEncoderLayer_46480136077658
// MI455X (gfx1250) — compile-verified
//
#include <hip/hip_runtime.h>
#include <hip/hip_bf16.h>

// Problem constants (from reference)
#define TOTAL   200000      // B*N
#define NF      128         // F
#define NFF     512         // FF
#define NE      6400000     // E
#define LN_EPS  1e-5f
#define LDS_STRIDE 132      // padded row stride (floats) for LDS tiles

typedef __attribute__((ext_vector_type(2))) float    v2f;
typedef __attribute__((ext_vector_type(8))) float    v8f;
typedef __attribute__((ext_vector_type(4))) unsigned v4u;
typedef __attribute__((ext_vector_type(8))) unsigned v8u;

// D = A(16x4) * B(4x16) + C, fp32 WMMA.
__device__ __forceinline__ v8f wmma_f32(v2f a, v2f b, v8f c) {
    return __builtin_amdgcn_wmma_f32_16x16x4_f32(false, a, false, b, (short)0, c,
                                                 false, false);
}

// generic shared-memory pointer -> 32-bit LDS byte offset (LDS aperture keeps
// the workgroup-relative offset in the low 32 bits of the generic address)
__device__ __forceinline__ unsigned lds_off(const void* p) {
    return (unsigned)(size_t)p;
}

// -------- CDNA5 async global->LDS (per-lane b128, tracked by ASYNCcnt) ------
__device__ __forceinline__ void async_b128(unsigned lds_byte, unsigned long long gaddr) {
    asm volatile("global_load_async_to_lds_b128 %0, %1, off"
                 :: "v"(lds_byte), "v"(gaddr) : "memory");
}
__device__ __forceinline__ void wait_async0() {
    asm volatile("s_wait_asynccnt 0x0" ::: "memory");
}

// -------- CDNA5 Tensor Data Mover: DMA a 128x128 f32 tile into LDS ----------
// Pads 4 DWORDs every 128 DWORDs (one row) -> LDS row stride of 132 floats,
// matching the WMMA B-fragment layout used below.  2-D descriptor, groups 2/3
// are NULL.  Issued by one wave; completion via TENSORcnt.
__device__ __forceinline__ void tdm_load_tile(const float* g, unsigned row_stride_f,
                                              unsigned lds_byte) {
    unsigned long long ga = (unsigned long long)(size_t)g;
    v4u g0 = { 1u,                                   // count=1, user descriptor
               lds_byte,                             // lds_addr
               (unsigned)ga,                         // global_addr[31:0]
               ((unsigned)(ga >> 32) & 0x1FFFFFFu) | 0x80000000u }; // addr[56:32] | type=2
    v8u g1;
    g1[0] = (2u << 16)      // data_size = 4 bytes
          | (1u << 20)      // pad_enable
          | (6u << 22)      // pad_interval: 128 DWORDs
          | (3u << 25);     // pad_amount : 4 DWORDs
    g1[1] = (row_stride_f & 0xFFFFu) << 16;          // tensor_dim0[15:0]
    g1[2] = (row_stride_f >> 16) | (128u << 16);     // tensor_dim0[31:16] | tensor_dim1[15:0]
    g1[3] = (128u << 16);                            // tensor_dim1[31:16]=0 | tile_dim0=128
    g1[4] = 128u;                                    // tile_dim1=128, tile_dim2=0
    g1[5] = row_stride_f;                            // tensor_dim0_stride[31:0]
    g1[6] = 0u;                                      // stride hi | dim1_stride lo
    g1[7] = 0u;
    asm volatile("tensor_load_to_lds %0, %1" :: "s"(g0), "s"(g1) : "memory");
}

// ---------------------------------------------------------------- degree ----
__global__ __launch_bounds__(256) void k_deg_init(float* __restrict__ deg) {
    int i = blockIdx.x * 256 + threadIdx.x;
    if (i < TOTAL) deg[i] = 1.0f;   // self-loop contribution
}

__global__ __launch_bounds__(256) void k_deg_count(const long long* __restrict__ ei,
                                                   float* __restrict__ deg) {
    int e = blockIdx.x * 256 + threadIdx.x;
    if (e < NE) atomicAdd(deg + (int)ei[(size_t)NE + e], 1.0f);
}

__global__ __launch_bounds__(256) void k_deg_rsqrt(float* __restrict__ deg) {
    int i = blockIdx.x * 256 + threadIdx.x;
    if (i < TOTAL) deg[i] = rsqrtf(deg[i]);   // deg >= 1 always (self-loops)
}

// ------------------------------------------------------- GCN linear (WMMA) --
// xw[TOTAL,128] = x[TOTAL,128] @ W[128,128]; W staged via async-to-LDS b128.
__global__ __launch_bounds__(256) void k_gcn_gemm(const float* __restrict__ x,
                                                  const float* __restrict__ W,
                                                  float* __restrict__ xw) {
    extern __shared__ float ldsW[];           // 128 x LDS_STRIDE floats
    const int tid  = threadIdx.x;
    const int wid  = tid >> 5;
    const int lane = tid & 31;
    const int mm   = lane & 15;
    const int hi   = lane >> 4;
    const int kh   = hi << 1;

    // async-stage W (row-major 128x128, 512B rows) into padded LDS:
    // 4096 b128 chunks, 16 per thread, no VGPR data round-trip.
    {
        unsigned lbase = lds_off(ldsW);
        unsigned long long gbase = (unsigned long long)(size_t)W;
#pragma unroll
        for (int i = 0; i < 16; ++i) {
            int chunk = tid + (i << 8);            // 0..4095
            int row = chunk >> 5;                  // 32 chunks per 512B row
            int cq  = chunk & 31;
            async_b128(lbase + (unsigned)row * (LDS_STRIDE * 4) + ((unsigned)cq << 4),
                       gbase + ((unsigned long long)row << 9) + ((unsigned)cq << 4));
        }
        wait_async0();
    }
    __syncthreads();

    const int rowBase = blockIdx.x * 128 + (wid << 4);
    if (rowBase >= TOTAL) return;             // whole-wave uniform (TOTAL % 16 == 0)

    v8f acc[8];
#pragma unroll
    for (int t = 0; t < 8; ++t) { v8f z = {}; acc[t] = z; }

    const float* arow = x + (size_t)(rowBase + mm) * NF;
    for (int k0 = 0; k0 < NF; k0 += 4) {
        v2f a = *(const v2f*)(arow + k0 + kh);
#pragma unroll
        for (int t = 0; t < 8; ++t) {
            v2f b;
            b.x = ldsW[(k0 + kh) * LDS_STRIDE + (t << 4) + mm];
            b.y = ldsW[(k0 + kh + 1) * LDS_STRIDE + (t << 4) + mm];
            acc[t] = wmma_f32(a, b, acc[t]);
        }
    }
#pragma unroll
    for (int t = 0; t < 8; ++t)
#pragma unroll
        for (int r = 0; r < 8; ++r)
            xw[(size_t)(rowBase + r + (hi << 3)) * NF + (t << 4) + mm] = acc[t][r];
}

// ------------------------------------------------------------- aggregation --
__global__ __launch_bounds__(256) void k_selfloop(const float* __restrict__ xw,
                                                  const float* __restrict__ dis,
                                                  float* __restrict__ agg) {
    int t = blockIdx.x * 256 + threadIdx.x;   // t < TOTAL*32
    int n = t >> 5, j = t & 31;
    float d = dis[n];
    float s = d * d;
    const float4 v = *(const float4*)(xw + (size_t)n * NF + j * 4);
    float4 o; o.x = v.x * s; o.y = v.y * s; o.z = v.z * s; o.w = v.w * s;
    *(float4*)(agg + (size_t)n * NF + j * 4) = o;
}

// one wave32 per edge: gather xw[row] (L2-resident), scale, atomic-add agg[col]
__global__ __launch_bounds__(256) void k_scatter(const long long* __restrict__ ei,
                                                 const float* __restrict__ xw,
                                                 const float* __restrict__ dis,
                                                 float* __restrict__ agg) {
    int e    = blockIdx.x * 8 + (threadIdx.x >> 5);
    int lane = threadIdx.x & 31;
    if (e >= NE) return;
    int row = (int)ei[e];
    int col = (int)ei[(size_t)NE + e];
    float nrm = dis[row] * dis[col];
    const float4 v = *(const float4*)(xw + (size_t)row * NF + lane * 4);
    float* dst = agg + (size_t)col * NF + lane * 4;
    atomicAdd(dst + 0, v.x * nrm);
    atomicAdd(dst + 1, v.y * nrm);
    atomicAdd(dst + 2, v.z * nrm);
    atomicAdd(dst + 3, v.w * nrm);
}

// ----------------------------------------------- bias + residual + LN1 ------
__global__ __launch_bounds__(256) void k_bias_res_ln1(const float* __restrict__ agg,
                                                      const float* __restrict__ x,
                                                      const float* __restrict__ bias,
                                                      const float* __restrict__ g,
                                                      const float* __restrict__ be,
                                                      float* __restrict__ h) {
    int row  = blockIdx.x * 8 + (threadIdx.x >> 5);
    int lane = threadIdx.x & 31;
    if (row >= TOTAL) return;
    const float4 a4 = *(const float4*)(agg + (size_t)row * NF + lane * 4);
    const float4 x4 = *(const float4*)(x + (size_t)row * NF + lane * 4);
    const float4 b4 = *(const float4*)(bias + lane * 4);
    float vx = a4.x + b4.x + x4.x;
    float vy = a4.y + b4.y + x4.y;
    float vz = a4.z + b4.z + x4.z;
    float vw = a4.w + b4.w + x4.w;
    float s = vx + vy + vz + vw;
    float q = vx * vx + vy * vy + vz * vz + vw * vw;
#pragma unroll
    for (int m = 16; m >= 1; m >>= 1) {
        s += __shfl_xor(s, m);
        q += __shfl_xor(q, m);
    }
    float mu  = s * (1.0f / 128.0f);
    float var = q * (1.0f / 128.0f) - mu * mu;
    float rs  = rsqrtf(var + LN_EPS);
    const float4 g4 = *(const float4*)(g + lane * 4);
    const float4 e4 = *(const float4*)(be + lane * 4);
    float4 o;
    o.x = (vx - mu) * rs * g4.x + e4.x;
    o.y = (vy - mu) * rs * g4.y + e4.y;
    o.z = (vz - mu) * rs * g4.z + e4.z;
    o.w = (vw - mu) * rs * g4.w + e4.w;
    *(float4*)(h + (size_t)row * NF + lane * 4) = o;
}

// ---------------------------- fused FFN (silu MLP) + residual + LN2 ---------
// per block: 128 rows; FF in 4 chunks of 128.  W1/W2 chunks DMA'd by the
// Tensor Data Mover with hardware row-padding (stride 132), hidden tile
// round-trips through LDS between the two WMMA GEMMs.
__global__ __launch_bounds__(256) void k_ffn_ln2(const float* __restrict__ h,
                                                 const float* __restrict__ W1,
                                                 const float* __restrict__ b1,
                                                 const float* __restrict__ W2,
                                                 const float* __restrict__ b2,
                                                 const float* __restrict__ g2,
                                                 const float* __restrict__ be2,
                                                 float* __restrict__ out) {
    extern __shared__ float lds[];
    float* ldsW = lds;                         // 128 x LDS_STRIDE
    float* ldsH = lds + NF * LDS_STRIDE;       // 128 x LDS_STRIDE
    const int tid  = threadIdx.x;
    const int wid  = tid >> 5;
    const int lane = tid & 31;
    const int mm   = lane & 15;
    const int hi   = lane >> 4;
    const int kh   = hi << 1;
    const int wBase = wid << 4;
    const int rowBase = blockIdx.x * 128 + wBase;
    const bool valid = rowBase < TOTAL;        // wave-uniform
    const unsigned ldsW_byte = lds_off(ldsW);

    v8f acc[8];
#pragma unroll
    for (int t = 0; t < 8; ++t) { v8f z = {}; acc[t] = z; }

    for (int c = 0; c < 4; ++c) {
        __syncthreads();                       // prior GEMM2 done with ldsW
        if (wid == 0) {                        // TDM: W1[:, c*128 .. +128)
            tdm_load_tile(W1 + ((size_t)c << 7), NFF, ldsW_byte);
            __builtin_amdgcn_s_wait_tensorcnt(0);
        }
        __syncthreads();

        v8f hid[8];
#pragma unroll
        for (int t = 0; t < 8; ++t) { v8f z = {}; hid[t] = z; }
        if (valid) {
            const float* arow = h + (size_t)(rowBase + mm) * NF;
            for (int k0 = 0; k0 < NF; k0 += 4) {
                v2f a = *(const v2f*)(arow + k0 + kh);
#pragma unroll
                for (int t = 0; t < 8; ++t) {
                    v2f b;
                    b.x = ldsW[(k0 + kh) * LDS_STRIDE + (t << 4) + mm];
                    b.y = ldsW[(k0 + kh + 1) * LDS_STRIDE + (t << 4) + mm];
                    hid[t] = wmma_f32(a, b, hid[t]);
                }
            }
            // +b1, SiLU, write hidden tile (own 16 rows only) to LDS
#pragma unroll
            for (int t = 0; t < 8; ++t) {
                float b1v = b1[(c << 7) + (t << 4) + mm];
#pragma unroll
                for (int r = 0; r < 8; ++r) {
                    float xv = hid[t][r] + b1v;
                    ldsH[(wBase + r + (hi << 3)) * LDS_STRIDE + (t << 4) + mm] =
                        xv / (1.0f + __expf(-xv));
                }
            }
        }
        __syncthreads();                       // GEMM1 done with ldsW
        if (wid == 0) {                        // TDM: W2[c*128 .. +128, :]
            tdm_load_tile(W2 + (((size_t)c << 7) * NF), NF, ldsW_byte);
            __builtin_amdgcn_s_wait_tensorcnt(0);
        }
        __syncthreads();

        if (valid) {
            for (int k0 = 0; k0 < NF; k0 += 4) {
                v2f a = *(const v2f*)(ldsH + (wBase + mm) * LDS_STRIDE + k0 + kh);
#pragma unroll
                for (int t = 0; t < 8; ++t) {
                    v2f b;
                    b.x = ldsW[(k0 + kh) * LDS_STRIDE + (t << 4) + mm];
                    b.y = ldsW[(k0 + kh + 1) * LDS_STRIDE + (t << 4) + mm];
                    acc[t] = wmma_f32(a, b, acc[t]);
                }
            }
        }
    }
    if (!valid) return;

    // epilogue: +b2, +residual, LN2
    float g2v[8], e2v[8];
#pragma unroll
    for (int t = 0; t < 8; ++t) {
        int col = (t << 4) + mm;
        float b2v = b2[col];
        g2v[t] = g2[col];
        e2v[t] = be2[col];
#pragma unroll
        for (int r = 0; r < 8; ++r)
            acc[t][r] += b2v + h[(size_t)(rowBase + r + (hi << 3)) * NF + col];
    }
#pragma unroll
    for (int r = 0; r < 8; ++r) {
        float s = 0.0f, q = 0.0f;
#pragma unroll
        for (int t = 0; t < 8; ++t) { float v = acc[t][r]; s += v; q += v * v; }
        s += __shfl_xor(s, 1);  q += __shfl_xor(q, 1);
        s += __shfl_xor(s, 2);  q += __shfl_xor(q, 2);
        s += __shfl_xor(s, 4);  q += __shfl_xor(q, 4);
        s += __shfl_xor(s, 8);  q += __shfl_xor(q, 8);
        float mu  = s * (1.0f / 128.0f);
        float var = q * (1.0f / 128.0f) - mu * mu;
        float rs  = rsqrtf(var + LN_EPS);
        size_t rowoff = (size_t)(rowBase + r + (hi << 3)) * NF;
#pragma unroll
        for (int t = 0; t < 8; ++t)
            out[rowoff + (t << 4) + mm] = (acc[t][r] - mu) * rs * g2v[t] + e2v[t];
    }
}

// ---------------------------------------------------------------------------
extern "C" void kernel_launch(void* const* d_in, const int* in_sizes, int n_in,
                              void* d_out, int out_size, void* d_ws, size_t ws_size,
                              hipStream_t stream) {
    const float*     x   = (const float*)d_in[0];
    const long long* ei  = (const long long*)d_in[1];
    const float*     Wg  = (const float*)d_in[2];
    const float*     bg  = (const float*)d_in[3];
    const float*     W1  = (const float*)d_in[4];
    const float*     b1  = (const float*)d_in[5];
    const float*     W2  = (const float*)d_in[6];
    const float*     b2  = (const float*)d_in[7];
    const float*     g1  = (const float*)d_in[8];
    const float*     be1 = (const float*)d_in[9];
    const float*     g2  = (const float*)d_in[10];
    const float*     be2 = (const float*)d_in[11];
    float* out = (float*)d_out;

    // workspace: dis[TOTAL] | xw[TOTAL*128] | agg[TOTAL*128]; h reuses xw
    float* ws  = (float*)d_ws;
    float* dis = ws;
    float* xw  = ws + 200704;                  // TOTAL rounded up, 16B aligned
    float* agg = xw + (size_t)TOTAL * NF;
    float* h   = xw;                           // safe reuse after scatter

    const size_t ldsA = (size_t)NF * LDS_STRIDE * sizeof(float);      //  ~66 KB
    const size_t ldsB = 2 * ldsA;                                     // ~132 KB

    k_deg_init  <<<(TOTAL + 255) / 256, 256, 0, stream>>>(dis);
    k_deg_count <<<NE / 256,            256, 0, stream>>>(ei, dis);
    k_deg_rsqrt <<<(TOTAL + 255) / 256, 256, 0, stream>>>(dis);

    k_gcn_gemm  <<<(TOTAL + 127) / 128, 256, ldsA, stream>>>(x, Wg, xw);

    k_selfloop  <<<(TOTAL * 32) / 256,  256, 0, stream>>>(xw, dis, agg);
    k_scatter   <<<NE / 8,              256, 0, stream>>>(ei, xw, dis, agg);

    k_bias_res_ln1 <<<TOTAL / 8, 256, 0, stream>>>(agg, x, bg, g1, be1, h);

    k_ffn_ln2   <<<(TOTAL + 127) / 128, 256, ldsB, stream>>>(h, W1, b1, W2, b2,
                                                             g2, be2, out);
}